// VisualAttention_33878702031528
// MI455X (gfx1250) — compile-verified
//
#include <hip/hip_runtime.h>

typedef float v2f __attribute__((ext_vector_type(2)));
typedef float v8f __attribute__((ext_vector_type(8)));

#define SQd 2048
#define Bd  4
#define Ed  1024
#define Hd  16
#define DHd 64
#define Rd  16

// ---- CDNA5 async global->LDS copy (ASYNCcnt path), 16B per lane ----
__device__ __forceinline__ void async_copy_b128(float* lds, const float* g) {
  asm volatile("global_load_async_to_lds_b128 %0, %1, off"
               :
               : "v"((unsigned int)(uintptr_t)lds),
                 "v"((unsigned long long)(uintptr_t)g)
               : "memory");
}
__device__ __forceinline__ void wait_async0() {
  asm volatile("s_wait_asynccnt 0" ::: "memory");
}

// ---------------- Kernel A: fold LoRA into weights: Weff = W + s * (lb @ la) ----------------
__global__ void fuse_weights_kernel(const float* __restrict__ w,
                                    const float* __restrict__ la,   // [R, K]
                                    const float* __restrict__ lb,   // [N, R]
                                    const float* __restrict__ sc,   // [1]
                                    float* __restrict__ weff,
                                    int N, int Kd) {
  int idx = blockIdx.x * blockDim.x + threadIdx.x;
  if (idx >= N * Kd) return;
  int n = idx / Kd, k = idx - n * Kd;
  float acc = 0.f;
#pragma unroll
  for (int r = 0; r < Rd; ++r) acc += lb[n * Rd + r] * la[r * Kd + k];
  weff[idx] = w[idx] + sc[0] * acc;
}

// ---------------- Kernel B/D: C = A[MxK] @ W[NxK]^T + bias, fp32 WMMA 16x16x4 ----------------
// 8 waves as 2(M) x 4(N); 64x64 per wave (4x4 WMMA tiles) -> 16 WMMA per 8 frag loads.
#define BM  128
#define BN  256
#define BKt 16
#define LDK 20   // BK + 4 pad: 16B-aligned float4 rows, conflict-free fragment reads

__global__ __launch_bounds__(256) void gemm_wmma_kernel(
    const float* __restrict__ A, const float* __restrict__ W,
    const float* __restrict__ bias,
    float* __restrict__ Cq, float* __restrict__ Ck, float* __restrict__ Cv,
    float* __restrict__ Cplain, int N, int Kd, int mode) {
  __shared__ float sA[BM * LDK];
  __shared__ float sB[BN * LDK];
  const int tid = threadIdx.x;
  const int lane = tid & 31;
  const int wv = tid >> 5;
  const int wm = wv >> 2;              // 0..1
  const int wn = wv & 3;               // 0..3
  const int hlf = lane >> 4, l16 = lane & 15;
  const int bm = blockIdx.x * BM;
  const int bn = blockIdx.y * BN;

  const v8f z8 = {0.f, 0.f, 0.f, 0.f, 0.f, 0.f, 0.f, 0.f};
  v8f acc[4][4];
#pragma unroll
  for (int i = 0; i < 4; ++i)
#pragma unroll
    for (int j = 0; j < 4; ++j) acc[i][j] = z8;

  for (int k0 = 0; k0 < Kd; k0 += BKt) {
    // A tile: 128x16 floats = 512 float4; 2 per thread
#pragma unroll
    for (int i = 0; i < 2; ++i) {
      int t = tid + i * 256;
      int row = t >> 2;
      int c4 = (t & 3) << 2;
      *reinterpret_cast<float4*>(&sA[row * LDK + c4]) =
          *reinterpret_cast<const float4*>(A + (size_t)(bm + row) * Kd + k0 + c4);
    }
    // W tile: 256x16 floats = 1024 float4; 4 per thread
#pragma unroll
    for (int i = 0; i < 4; ++i) {
      int t = tid + i * 256;
      int row = t >> 2;
      int c4 = (t & 3) << 2;
      *reinterpret_cast<float4*>(&sB[row * LDK + c4]) =
          *reinterpret_cast<const float4*>(W + (size_t)(bn + row) * Kd + k0 + c4);
    }
    __syncthreads();
#pragma unroll
    for (int kk = 0; kk < BKt; kk += 4) {
      const int kc = kk + hlf * 2;  // lanes 0-15: K=k0,k0+1 ; lanes 16-31: K=k0+2,k0+3
      v2f a[4], b[4];
#pragma unroll
      for (int i = 0; i < 4; ++i)
        a[i] = *reinterpret_cast<const v2f*>(&sA[(wm * 64 + i * 16 + l16) * LDK + kc]);
#pragma unroll
      for (int j = 0; j < 4; ++j)
        b[j] = *reinterpret_cast<const v2f*>(&sB[(wn * 64 + j * 16 + l16) * LDK + kc]);
#pragma unroll
      for (int i = 0; i < 4; ++i)
#pragma unroll
        for (int j = 0; j < 4; ++j)
          acc[i][j] = __builtin_amdgcn_wmma_f32_16x16x4_f32(
              false, a[i], false, b[j], (short)0, acc[i][j], false, false);
    }
    __syncthreads();
  }

  // Epilogue. C layout: VGPR r holds M=r (lanes 0-15) / M=r+8 (lanes 16-31), N=l16.
#pragma unroll
  for (int mi = 0; mi < 4; ++mi) {
#pragma unroll
    for (int ni = 0; ni < 4; ++ni) {
      const int rowb = bm + wm * 64 + mi * 16 + hlf * 8;
      const int col = bn + wn * 64 + ni * 16 + l16;
      const float bv = bias[col];
#pragma unroll
      for (int r = 0; r < 8; ++r) {
        const int row = rowb + r;                 // row = s*B + b
        const float val = acc[mi][ni][r] + bv;
        if (mode == 1) {
          Cplain[(size_t)row * N + col] = val;
        } else {
          // mixed col -> (h, c) with c in [0,192); scatter to [b,h,sq,dh]
          const int h = col / 192;
          const int c = col - h * 192;
          const int s = row >> 2;
          const int b = row & 3;
          const size_t base = ((size_t)(b * Hd + h) * SQd + s) * DHd;
          if (c < 64)       Cq[base + c] = val * 0.125f;       // 1/sqrt(64)
          else if (c < 128) Ck[base + (c - 64)] = val;
          else              Cv[base + (c - 128)] = val;
        }
      }
    }
  }
}

// ---------------- Kernel C: flash attention per (b,h) ----------------
#define FQ  128   // q rows per block (8 waves x 16)
#define FK  64    // key tile
#define LDV 68    // DH + 4 pad

__global__ __launch_bounds__(256) void attn_flash_kernel(
    const float* __restrict__ Qg, const float* __restrict__ Kg,
    const float* __restrict__ Vg, const float* __restrict__ Mg,
    float* __restrict__ ctx) {
  __shared__ float sK[FK * LDV];
  __shared__ float sV[FK * LDV];
  __shared__ float sP[8 * 16 * LDV];
  const int tid = threadIdx.x;
  const int lane = tid & 31;
  const int wv = tid >> 5;
  const int hlf = lane >> 4, l16 = lane & 15;
  const int bh = blockIdx.y;                 // b*H + h
  const int qbase = blockIdx.x * FQ;
  const float* Qb = Qg + (size_t)bh * SQd * DHd;
  const float* Kb = Kg + (size_t)bh * SQd * DHd;
  const float* Vb = Vg + (size_t)bh * SQd * DHd;
  const float* Mb = Mg + (size_t)bh * SQd * SQd;

  // Preload Q A-fragments (16 k-steps of 4 over DH=64) into registers
  const int qrow = qbase + wv * 16 + l16;
  v2f qf[16];
#pragma unroll
  for (int kk = 0; kk < 16; ++kk)
    qf[kk] = *reinterpret_cast<const v2f*>(Qb + (size_t)qrow * DHd + kk * 4 + hlf * 2);

  const v8f z8 = {0.f, 0.f, 0.f, 0.f, 0.f, 0.f, 0.f, 0.f};
  v8f o[4] = {z8, z8, z8, z8};
  float mrow[8], lrow[8];
#pragma unroll
  for (int r = 0; r < 8; ++r) { mrow[r] = -1e30f; lrow[r] = 0.f; }

  float* myP = &sP[wv * 16 * LDV];

  for (int kb = 0; kb < SQd; kb += FK) {
    // Async K/V tile load straight into LDS (ASYNCcnt path): 16B per lane per issue
#pragma unroll
    for (int i = 0; i < 4; ++i) {
      int t = tid + i * 256;
      int row = t >> 4;
      int c4 = (t & 15) << 2;
      async_copy_b128(&sK[row * LDV + c4], Kb + (size_t)(kb + row) * DHd + c4);
      async_copy_b128(&sV[row * LDV + c4], Vb + (size_t)(kb + row) * DHd + c4);
    }
    wait_async0();
    __syncthreads();

    // S = Q @ K^T over DH (B-frag lane n holds K-row n, dh pair)
    v8f s[4] = {z8, z8, z8, z8};
#pragma unroll
    for (int j = 0; j < 4; ++j) {
#pragma unroll
      for (int kk = 0; kk < 16; ++kk) {
        v2f b = *reinterpret_cast<const v2f*>(&sK[(j * 16 + l16) * LDV + kk * 4 + hlf * 2]);
        s[j] = __builtin_amdgcn_wmma_f32_16x16x4_f32(false, qf[kk], false, b, (short)0, s[j], false, false);
      }
    }

    // + attn_mask tile (the dominant HBM stream: prefetch next tile)
    const float* mp0 = Mb + (size_t)(qbase + wv * 16 + hlf * 8) * SQd + kb;
    if (kb + FK < SQd) __builtin_prefetch(mp0 + FK, 0, 0);
#pragma unroll
    for (int j = 0; j < 4; ++j)
#pragma unroll
      for (int r = 0; r < 8; ++r)
        s[j][r] += mp0[(size_t)r * SQd + j * 16 + l16];

    // online softmax: row reductions within 16-lane halves
#pragma unroll
    for (int r = 0; r < 8; ++r) {
      float v = fmaxf(fmaxf(s[0][r], s[1][r]), fmaxf(s[2][r], s[3][r]));
      v = fmaxf(v, __shfl_xor(v, 1, 32));
      v = fmaxf(v, __shfl_xor(v, 2, 32));
      v = fmaxf(v, __shfl_xor(v, 4, 32));
      v = fmaxf(v, __shfl_xor(v, 8, 32));
      const float nm = fmaxf(mrow[r], v);
      const float corr = __expf(mrow[r] - nm);
      mrow[r] = nm;
      float rs = 0.f;
#pragma unroll
      for (int j = 0; j < 4; ++j) {
        float p = __expf(s[j][r] - nm);
        s[j][r] = p;
        rs += p;
      }
      rs += __shfl_xor(rs, 1, 32);
      rs += __shfl_xor(rs, 2, 32);
      rs += __shfl_xor(rs, 4, 32);
      rs += __shfl_xor(rs, 8, 32);
      lrow[r] = lrow[r] * corr + rs;
#pragma unroll
      for (int n = 0; n < 4; ++n) o[n][r] *= corr;
    }

    // stage P (C layout -> LDS row-major) for PV A-fragments
#pragma unroll
    for (int j = 0; j < 4; ++j)
#pragma unroll
      for (int r = 0; r < 8; ++r)
        myP[(r + hlf * 8) * LDV + j * 16 + l16] = s[j][r];
    __syncthreads();

    // O += P @ V  (B-frag lane n holds V column n, key pair)
#pragma unroll
    for (int kk = 0; kk < 16; ++kk) {
      v2f a = *reinterpret_cast<const v2f*>(&myP[l16 * LDV + kk * 4 + hlf * 2]);
      const int krow = kk * 4 + hlf * 2;
#pragma unroll
      for (int n = 0; n < 4; ++n) {
        v2f b;
        b.x = sV[krow * LDV + n * 16 + l16];
        b.y = sV[(krow + 1) * LDV + n * 16 + l16];
        o[n] = __builtin_amdgcn_wmma_f32_16x16x4_f32(false, a, false, b, (short)0, o[n], false, false);
      }
    }
    __syncthreads();
  }

  // normalize, write ctx[s][b][h*64 + d]
  const int b = bh / Hd, h = bh % Hd;
#pragma unroll
  for (int n = 0; n < 4; ++n)
#pragma unroll
    for (int r = 0; r < 8; ++r) {
      const int srow = qbase + wv * 16 + hlf * 8 + r;
      ctx[((size_t)srow * Bd + b) * Ed + h * DHd + n * 16 + l16] = o[n][r] / lrow[r];
    }
}

extern "C" void kernel_launch(void* const* d_in, const int* in_sizes, int n_in,
                              void* d_out, int out_size, void* d_ws, size_t ws_size,
                              hipStream_t stream) {
  (void)in_sizes; (void)n_in; (void)out_size; (void)ws_size;
  const float* query  = (const float*)d_in[0];
  const float* mask   = (const float*)d_in[1];
  const float* w_in   = (const float*)d_in[2];
  const float* b_in   = (const float*)d_in[3];
  const float* la_in  = (const float*)d_in[4];
  const float* lb_in  = (const float*)d_in[5];
  const float* sc_in  = (const float*)d_in[6];
  const float* w_out  = (const float*)d_in[7];
  const float* b_out  = (const float*)d_in[8];
  const float* la_out = (const float*)d_in[9];
  const float* lb_out = (const float*)d_in[10];
  const float* sc_out = (const float*)d_in[11];
  float* out = (float*)d_out;

  float* ws = (float*)d_ws;
  float* weff_in  = ws;                                   // 3E*E
  float* weff_out = weff_in  + (size_t)3 * Ed * Ed;       // E*E
  float* Qb       = weff_out + (size_t)Ed * Ed;           // [BH][SQ][DH]
  float* Kb       = Qb + (size_t)SQd * Bd * Ed;
  float* Vb       = Kb + (size_t)SQd * Bd * Ed;
  float* ctx      = Vb + (size_t)SQd * Bd * Ed;           // [SQ*B][E]

  const int M = SQd * Bd;       // 8192
  const int N3 = 3 * Ed;        // 3072

  fuse_weights_kernel<<<(N3 * Ed + 255) / 256, 256, 0, stream>>>(
      w_in, la_in, lb_in, sc_in, weff_in, N3, Ed);
  fuse_weights_kernel<<<(Ed * Ed + 255) / 256, 256, 0, stream>>>(
      w_out, la_out, lb_out, sc_out, weff_out, Ed, Ed);

  gemm_wmma_kernel<<<dim3(M / BM, N3 / BN), 256, 0, stream>>>(
      query, weff_in, b_in, Qb, Kb, Vb, nullptr, N3, Ed, 0);

  attn_flash_kernel<<<dim3(SQd / FQ, Bd * Hd), 256, 0, stream>>>(
      Qb, Kb, Vb, mask, ctx);

  gemm_wmma_kernel<<<dim3(M / BM, Ed / BN), 256, 0, stream>>>(
      ctx, weff_out, b_out, nullptr, nullptr, nullptr, out, Ed, Ed, 1);
}